// GinAgg_34737695490534
// MI455X (gfx1250) — compile-verified
//
#include <hip/hip_runtime.h>
#include <hip/hip_bf16.h>
#include <stddef.h>

#define N_EDGES 640000
#define N_NODES 20000
#define DIM     128
#define LDS_STRIDE 132   // pad 128 -> 132 so column reads (stride%64==4) hit distinct banks

typedef float v2f __attribute__((ext_vector_type(2)));
typedef float v4f __attribute__((ext_vector_type(4)));
typedef float v8f __attribute__((ext_vector_type(8)));

// ---------------------------------------------------------------------------
// Kernel 1: zero the aggregation accumulator in workspace (must re-zero every
// launch: harness does not re-poison between replays, and scatter accumulates).
// Default (RT) stores so agg lines are established in L2 and stay resident.
// ---------------------------------------------------------------------------
__global__ __launch_bounds__(256) void zero_agg_kernel(v4f* __restrict__ agg4) {
    const int i = blockIdx.x * blockDim.x + threadIdx.x;   // N_NODES*DIM/4 = 640000 v4f
    if (i < N_NODES * DIM / 4) {
        agg4[i] = (v4f){0.f, 0.f, 0.f, 0.f};
    }
}

// ---------------------------------------------------------------------------
// Kernel 2: segment_sum. One wave32 per edge; each lane moves 4 consecutive
// floats and issues 4 L2-resident global_atomic_add_f32.
// The 327 MB stream of x is the roofline term (~14 us @ 23.3 TB/s); it is
// loaded NON-TEMPORAL so it cannot evict the 10.24 MB agg accumulator (which
// must stay L2-hot for the 82M f32 atomics) or the 128 KB of MLP weights.
// ---------------------------------------------------------------------------
__global__ __launch_bounds__(256) void scatter_sum_kernel(const float* __restrict__ x,
                                                          const int*   __restrict__ index,
                                                          float*       __restrict__ agg) {
    const int wavesPerBlock = 256 / 32;
    const int edge = blockIdx.x * wavesPerBlock + (threadIdx.x >> 5);
    const int lane = threadIdx.x & 31;
    if (edge >= N_EDGES) return;

    const int node = index[edge];

    const v4f* xrow = (const v4f*)(x + (size_t)edge * DIM);
    // Prefetch the next edge's row while this one is in flight (global_prefetch_b8).
    if (edge + wavesPerBlock < N_EDGES) {
        __builtin_prefetch(x + (size_t)(edge + wavesPerBlock) * DIM + lane * 4, 0, 0);
    }
    const v4f v = __builtin_nontemporal_load(xrow + lane);   // TH=NT global_load_b128

    float* dst = agg + (size_t)node * DIM + lane * 4;
    atomicAdd(dst + 0, v.x);
    atomicAdd(dst + 1, v.y);
    atomicAdd(dst + 2, v.z);
    atomicAdd(dst + 3, v.w);
}

// ---------------------------------------------------------------------------
// Kernel 3: fused 2-layer MLP using V_WMMA_F32_16X16X4_F32.
// Workgroup = 16 node rows; 8 waves each own a 16-wide N tile (8*16 = 128).
// Layer-1 output tile (16x128) == layer-2 K panel -> ReLU through LDS, no
// global traffic for activations. 20000 % 16 == 0, so no tails; EXEC all-1s.
// agg/W reads use default RT (L2-hot); final out stores are non-temporal
// (written once, never re-read).
// ---------------------------------------------------------------------------
__global__ __launch_bounds__(256) void mlp_wmma_kernel(const float* __restrict__ agg,
                                                       const float* __restrict__ W1,
                                                       const float* __restrict__ b1,
                                                       const float* __restrict__ W2,
                                                       const float* __restrict__ b2,
                                                       float*       __restrict__ out) {
    __shared__ float Atile[16 * LDS_STRIDE];  // agg rows for this block
    __shared__ float Htile[16 * LDS_STRIDE];  // relu(layer1) rows

    const int row0  = blockIdx.x * 16;
    const int tid   = threadIdx.x;
    const int wave  = tid >> 5;        // 0..7 (wave32)
    const int lane  = tid & 31;
    const int ncol0 = wave * 16;       // N tile for this wave

    // Cooperative load of the 16x128 A tile (coalesced loads, L2 hits).
    for (int i = tid; i < 16 * DIM; i += 256) {
        const int r = i >> 7;
        const int c = i & (DIM - 1);
        Atile[r * LDS_STRIDE + c] = agg[(size_t)(row0 + r) * DIM + c];
    }
    __syncthreads();

    // Per-lane fragment coordinates (ISA 7.12.2 layouts):
    //   A 16x4 f32: lanes 0-15 -> rows, K=0,1; lanes 16-31 -> K=2,3.
    //   C/D 16x16: VGPR i = row (lane<16 ? i : 8+i), col = lane%16.
    const int mrow  = lane & 15;
    const int koff  = (lane >> 4) * 2;      // 0 or 2
    const int ncol  = ncol0 + (lane & 15);
    const int rbase = (lane >> 4) * 8;

    // ---- Layer 1: h = relu(agg @ W1 + b1) ----
    v8f acc = {};
    for (int k = 0; k < DIM; k += 4) {
        v2f a, b;
        a.x = Atile[mrow * LDS_STRIDE + k + koff + 0];
        a.y = Atile[mrow * LDS_STRIDE + k + koff + 1];
        b.x = W1[(size_t)(k + koff + 0) * DIM + ncol];
        b.y = W1[(size_t)(k + koff + 1) * DIM + ncol];
        acc = __builtin_amdgcn_wmma_f32_16x16x4_f32(
            /*neg_a=*/false, a, /*neg_b=*/false, b,
            /*c_mod=*/(short)0, acc, /*reuse_a=*/false, /*reuse_b=*/false);
    }
    {
        const float bias = b1[ncol];
#pragma unroll
        for (int i = 0; i < 8; ++i) {
            float v = acc[i] + bias;
            v = v > 0.f ? v : 0.f;
            Htile[(rbase + i) * LDS_STRIDE + ncol] = v;
        }
    }
    __syncthreads();

    // ---- Layer 2: out = h @ W2 + b2 ----
    v8f acc2 = {};
    for (int k = 0; k < DIM; k += 4) {
        v2f a, b;
        a.x = Htile[mrow * LDS_STRIDE + k + koff + 0];
        a.y = Htile[mrow * LDS_STRIDE + k + koff + 1];
        b.x = W2[(size_t)(k + koff + 0) * DIM + ncol];
        b.y = W2[(size_t)(k + koff + 1) * DIM + ncol];
        acc2 = __builtin_amdgcn_wmma_f32_16x16x4_f32(
            false, a, false, b, (short)0, acc2, false, false);
    }
    {
        const float bias = b2[ncol];
#pragma unroll
        for (int i = 0; i < 8; ++i) {
            __builtin_nontemporal_store(acc2[i] + bias,
                                        &out[(size_t)(row0 + rbase + i) * DIM + ncol]);
        }
    }
}

// ---------------------------------------------------------------------------
// Launch: zero -> scatter -> fused MLP, all on `stream`.
// ---------------------------------------------------------------------------
extern "C" void kernel_launch(void* const* d_in, const int* in_sizes, int n_in,
                              void* d_out, int out_size, void* d_ws, size_t ws_size,
                              hipStream_t stream) {
    const float* x     = (const float*)d_in[0];
    const int*   index = (const int*)  d_in[1];   // int64 request downcast to i32 (x64 off)
    const float* W1    = (const float*)d_in[2];
    const float* b1    = (const float*)d_in[3];
    const float* W2    = (const float*)d_in[4];
    const float* b2    = (const float*)d_in[5];
    float*       out   = (float*)d_out;

    float* agg = (float*)d_ws;                    // N_NODES*DIM*4 = 10.24 MB scratch

    // 1) zero accumulator (2.56M floats = 640K v4f)
    {
        const int total4 = N_NODES * DIM / 4;
        const int blocks = (total4 + 255) / 256;
        zero_agg_kernel<<<blocks, 256, 0, stream>>>((v4f*)agg);
    }
    // 2) scatter-sum: 8 edges (waves) per 256-thread block
    {
        const int blocks = (N_EDGES + 7) / 8;
        scatter_sum_kernel<<<blocks, 256, 0, stream>>>(x, index, agg);
    }
    // 3) fused WMMA MLP: one block per 16-row node tile
    {
        const int blocks = N_NODES / 16;          // 20000 % 16 == 0
        mlp_wmma_kernel<<<blocks, 256, 0, stream>>>(agg, W1, b1, W2, b2, out);
    }
}